// AngleTensor_69767448756825
// MI455X (gfx1250) — compile-verified
//
#include <hip/hip_runtime.h>
#include <math.h>

typedef __attribute__((ext_vector_type(2))) float v2f;
typedef __attribute__((ext_vector_type(8))) float v8f;

#define NPTS 128
// EPS = 1e-10 on the product d_j*d_k separates as 1e-5 per factor
#define DSEP 1e-5f

// One block per (b, i): computes angles[b, i, :, :] (128x128).
// Normalized diff rows U[j] = (p_j - p_i)/d_ij are staged in LDS (K padded 3->4),
// then G = U * U^T via V_WMMA_F32_16X16X4_F32 gives cosines directly.
__global__ __launch_bounds__(256) void angle_tensor_kernel(
    const float* __restrict__ pos,    // (8, 128, 3)
    const float* __restrict__ dist,   // (8, 128, 128)
    float* __restrict__ out)          // (8, 128, 128, 128)
{
    __shared__ float s_udiff[NPTS * 4];  // normalized diff rows, 4th comp = 0

    const int bi  = blockIdx.x;          // 0..1023
    const int b   = bi >> 7;
    const int i   = bi & 127;
    const int tid = threadIdx.x;

    const float* posb = pos + (size_t)b * NPTS * 3;
    const float px = posb[i * 3 + 0];
    const float py = posb[i * 3 + 1];
    const float pz = posb[i * 3 + 2];

    if (tid < NPTS) {
        const int j = tid;
        const float dj = dist[((size_t)b * NPTS + i) * NPTS + j];
        const float rj = (dj < DSEP) ? 1.0f : (1.0f / dj);  // mp<EPS -> identity scale
        s_udiff[j * 4 + 0] = (posb[j * 3 + 0] - px) * rj;
        s_udiff[j * 4 + 1] = (posb[j * 3 + 1] - py) * rj;
        s_udiff[j * 4 + 2] = (posb[j * 3 + 2] - pz) * rj;
        s_udiff[j * 4 + 3] = 0.0f;
    }
    __syncthreads();

    const int wave = tid >> 5;           // 0..7 -> row tile jm
    const int lane = tid & 31;
    const int half = lane >> 4;          // A/B: 0 -> K=0,1 ; 1 -> K=2,3
    const int r    = lane & 15;

    // A-matrix (16x4 f32): lane<16 -> row r, K=0,1 ; lane>=16 -> row r, K=2,3
    const int jt = wave * 16;
    v2f amat;
    amat.x = s_udiff[(jt + r) * 4 + half * 2 + 0];
    amat.y = s_udiff[(jt + r) * 4 + half * 2 + 1];

    float* outbase = out + (size_t)bi * NPTS * NPTS;

    #pragma unroll
    for (int kn = 0; kn < 8; ++kn) {
        const int kt = kn * 16;
        // B-matrix (4x16 f32) = U^T: identical per-lane layout, column tile kt
        v2f bmat;
        bmat.x = s_udiff[(kt + r) * 4 + half * 2 + 0];
        bmat.y = s_udiff[(kt + r) * 4 + half * 2 + 1];

        v8f c = {};
        // 8 args: (neg_a, A, neg_b, B, c_mod, C, reuse_a, reuse_b)
        c = __builtin_amdgcn_wmma_f32_16x16x4_f32(
            false, amat, false, bmat, (short)0, c, false, false);

        const int kcol = kt + r;

        #pragma unroll
        for (int v = 0; v < 8; ++v) {
            const int j = jt + v + half * 8;   // C/D layout: lanes 16-31 hold M = v+8
            float cs = fminf(1.0f, fmaxf(-1.0f, c[v]));
            const bool masked = (i == j) | (i == kcol) | (j == kcol);
            const float ang = masked ? 0.0f : acosf(cs);
            outbase[(size_t)j * NPTS + kcol] = ang;
        }
    }
}

extern "C" void kernel_launch(void* const* d_in, const int* in_sizes, int n_in,
                              void* d_out, int out_size, void* d_ws, size_t ws_size,
                              hipStream_t stream) {
    const float* positions = (const float*)d_in[0];   // (8,128,3)
    const float* dist      = (const float*)d_in[1];   // (8,128,128)
    float* out             = (float*)d_out;           // (8,128,128,128)

    const int B = 8;
    dim3 grid(B * NPTS);   // one block per (b, i)
    dim3 block(256);       // 8 waves of 32
    angle_tensor_kernel<<<grid, block, 0, stream>>>(positions, dist, out);
}